// LinearLLM_45689862095163
// MI455X (gfx1250) — compile-verified
//
#include <hip/hip_runtime.h>

// CDNA5 WMMA operand vector types (f32 16x16x4: A,B = 2 VGPRs; C/D = 8 VGPRs)
typedef __attribute__((ext_vector_type(2))) float v2f;
typedef __attribute__((ext_vector_type(8))) float v8f;

#define SDIM 128

// out[b,t,v] = bias[t,v] + sum_{s>=t} weight[s, src[b,s], t, v]
// One wave32 per (b,t). s-reduction in chunks of 4 via V_WMMA_F32_16X16X4_F32
// with A = ones(16x4): D[m,n] = sum_k B[k,n] (replicated over m).
__global__ __launch_bounds__(32) void linllm_wmma_kernel(
    const int* __restrict__ src, const float* __restrict__ weight,
    const float* __restrict__ bias, float* __restrict__ out)
{
    const int bt   = blockIdx.x;      // 0 .. B*T-1
    const int b    = bt >> 7;
    const int t    = bt & 127;
    const int lane = threadIdx.x;     // 0..31
    const int h    = lane >> 4;       // half-wave id -> B-matrix K parity
    const int n    = lane & 15;       // column within a 16-wide v tile

    const int* srow = src + b * SDIM;

    // ---- head peel: s in [t, align4(t)) with plain VALU adds ----
    const int s0 = (t + 3) & ~3;
    float head[8];
#pragma unroll
    for (int p = 0; p < 8; ++p) head[p] = 0.0f;
    for (int s = t; s < s0; ++s) {
        const int w = srow[s];                      // wave-uniform
        const float* r = weight + ((((long)s * 128 + w) * 128 + t) * 128);
#pragma unroll
        for (int p = 0; p < 8; ++p) head[p] += r[p * 16 + n];
    }

    // ---- main loop: 4 s-rows per WMMA step, 8 v-tiles of 16 ----
    v8f C[8] = {};                                  // zero accumulators
    const v2f A = {1.0f, 1.0f};                     // ones(16x4) A-operand

    const int4* srow4 = (const int4*)srow;          // 16B-aligned (b*512B)
    for (int s4 = s0; s4 < SDIM; s4 += 4) {
        const int4 w4 = srow4[s4 >> 2];             // uniform s_load_b128
        const int wlo = h ? w4.y : w4.x;            // gather idx for row s4+h
        const int whi = h ? w4.w : w4.z;            // gather idx for row s4+2+h
        const float* r0 = weight + ((((long)(s4 + h)     * 128 + wlo) * 128 + t) * 128);
        const float* r1 = weight + ((((long)(s4 + 2 + h) * 128 + whi) * 128 + t) * 128);
#pragma unroll
        for (int p = 0; p < 8; ++p) {
            // B layout: VGPR0 = row (s4+h), VGPR1 = row (s4+2+h); cols across lanes
            v2f Bv = { r0[p * 16 + n], r1[p * 16 + n] };
            C[p] = __builtin_amdgcn_wmma_f32_16x16x4_f32(
                       /*neg_a=*/false, A, /*neg_b=*/false, Bv,
                       /*c_mod=*/(short)0, C[p],
                       /*reuse_a=*/false, /*reuse_b=*/false);
        }
    }

    // D rows are identical (A is all-ones), so lane n / VGPR0 holds out[.,16p+n]
    if (lane < 16) {
        const long ob = ((long)b * 128 + t) * 128;
#pragma unroll
        for (int p = 0; p < 8; ++p)
            out[ob + p * 16 + n] = C[p][0] + head[p] + bias[t * 128 + p * 16 + n];
    }
}

extern "C" void kernel_launch(void* const* d_in, const int* in_sizes, int n_in,
                              void* d_out, int out_size, void* d_ws, size_t ws_size,
                              hipStream_t stream) {
    (void)in_sizes; (void)n_in; (void)out_size; (void)d_ws; (void)ws_size;
    const int*   src    = (const int*)d_in[0];    // [32,128] int32
    const float* weight = (const float*)d_in[1];  // [128,128,128,128] f32
    const float* bias   = (const float*)d_in[2];  // [128,128] f32
    // d_in[3] (mask) is known lower-triangular: handled analytically (s >= t).
    float* out = (float*)d_out;                   // [32,128,128] f32

    dim3 grid(32 * 128);   // one wave32 per (b, t)
    dim3 block(32);
    hipLaunchKernelGGL(linllm_wmma_kernel, grid, block, 0, stream,
                       src, weight, bias, out);
}